// MSDenseGrid_87591563035292
// MI455X (gfx1250) — compile-verified
//
#include <hip/hip_runtime.h>

// Multi-scale dense-grid trilinear sampler for MI455X (gfx1250).
//
// Strategy: counting-sort the 1M query points by 15-bit Morton key (32^3
// coarse cells), then gather in sorted order. Concurrently-running waves then
// touch a compact spatial slab of each grid whose footprint fits the 192 MB
// L2, converting ~18-145x per-cacheline reuse potential into actual hits:
// HBM traffic drops from ~6 GB (random order, line-amplified) toward the
// ~550 MB compulsory footprint of the three grids.
//
// Gathers use the SADDR+voffset form (uniform channel base + per-lane u32
// corner offset). Output and sorted-xyz streams are non-temporal so the L2
// stays dedicated to grid data.

typedef float f2 __attribute__((ext_vector_type(2)));
typedef float f4 __attribute__((ext_vector_type(4)));
typedef f2 f2u __attribute__((aligned(4)));   // z-pair at arbitrary dword offset

#define NCH 12
#define NBINS 32768          // 32^3 Morton bins
#define SCAN_THREADS 1024    // NBINS / 32 items per thread

// ---------------------------------------------------------------- sampling --
template <int S, bool NT>
__device__ __forceinline__ void sample_grid(const float* __restrict__ g,
                                            float x, float y, float z,
                                            float acc[NCH]) {
  const float s1 = (float)(S - 1);
  float px = fminf(fmaxf(x * s1, 0.0f), s1);
  float py = fminf(fmaxf(y * s1, 0.0f), s1);
  float pz = fminf(fmaxf(z * s1, 0.0f), s1);
  float fx0 = floorf(px), fy0 = floorf(py), fz0 = floorf(pz);
  int ix0 = (int)fx0; if (ix0 > S - 1) ix0 = S - 1;
  int iy0 = (int)fy0; if (iy0 > S - 1) iy0 = S - 1;
  int iz0 = (int)fz0; if (iz0 > S - 1) iz0 = S - 1;
  int ix1 = min(ix0 + 1, S - 1);
  int iy1 = min(iy0 + 1, S - 1);
  int izb = min(iz0, S - 2);          // always load pair (izb, izb+1)
  bool zdup = (iz0 != izb);           // iz0==S-1 border case

  float fx = px - fx0, fy = py - fy0, fz = pz - fz0;
  float wx0 = 1.0f - fx, wy0 = 1.0f - fy, wz0 = 1.0f - fz;
  float wxy[4] = {wx0 * wy0, wx0 * fy, fx * wy0, fx * fy};

  unsigned bx0 = (unsigned)ix0 * (S * S), bx1 = (unsigned)ix1 * (S * S);
  unsigned by0 = (unsigned)iy0 * S, by1 = (unsigned)iy1 * S;
  unsigned voff[4] = {bx0 + by0 + (unsigned)izb, bx0 + by1 + (unsigned)izb,
                      bx1 + by0 + (unsigned)izb, bx1 + by1 + (unsigned)izb};

  float wa[4], wb[4];
#pragma unroll
  for (int k = 0; k < 4; ++k) {
    float a = wxy[k] * wz0;
    float b = wxy[k] * fz;
    wa[k] = zdup ? 0.0f : a;          // fold border case into weights
    wb[k] = zdup ? (a + b) : b;
  }

#pragma unroll
  for (int c = 0; c < NCH; ++c) {
    const float* __restrict__ gc = g + (size_t)c * (size_t)(S * S * S);
#pragma unroll
    for (int k = 0; k < 4; ++k) {
      const f2u* p = (const f2u*)(gc + voff[k]);
      f2 v;
      if constexpr (NT) {
        v = __builtin_nontemporal_load(p);
      } else {
        v = *p;
      }
      acc[c] = fmaf(wa[k], v.x, fmaf(wb[k], v.y, acc[c]));
    }
  }
}

__device__ __forceinline__ void sample_all(float x, float y, float z,
                                           const float* __restrict__ g0,
                                           const float* __restrict__ g1,
                                           const float* __restrict__ g2,
                                           float acc[NCH], bool nt2) {
#pragma unroll
  for (int c = 0; c < NCH; ++c) acc[c] = 0.0f;
  if (nt2) {  // unsorted fallback: keep streaming grid2 out of L2
    sample_grid<96, false>(g0, x, y, z, acc);
    sample_grid<144, false>(g1, x, y, z, acc);
    sample_grid<192, true>(g2, x, y, z, acc);
  } else {    // sorted path: spatial locality -> let L2/WGP$ capture reuse
    sample_grid<96, false>(g0, x, y, z, acc);
    sample_grid<144, false>(g1, x, y, z, acc);
    sample_grid<192, false>(g2, x, y, z, acc);
  }
}

__device__ __forceinline__ void store_out(float* __restrict__ out, int n,
                                          const float acc[NCH]) {
  f4* o = (f4*)(out + (size_t)n * NCH);   // 48 B per row, 16 B aligned
  f4 o0 = {acc[0], acc[1], acc[2], acc[3]};
  f4 o1 = {acc[4], acc[5], acc[6], acc[7]};
  f4 o2 = {acc[8], acc[9], acc[10], acc[11]};
  __builtin_nontemporal_store(o0, o + 0);
  __builtin_nontemporal_store(o1, o + 1);
  __builtin_nontemporal_store(o2, o + 2);
}

// ------------------------------------------------------------- Morton sort --
__device__ __forceinline__ unsigned spread3(unsigned v) {
  v &= 0x3FF;
  v = (v | (v << 16)) & 0x030000FF;
  v = (v | (v << 8))  & 0x0300F00F;
  v = (v | (v << 4))  & 0x030C30C3;
  v = (v | (v << 2))  & 0x09249249;
  return v;
}

__device__ __forceinline__ unsigned bin_key(float x, float y, float z) {
  int cx = max(0, min(31, (int)floorf(x * 32.0f)));
  int cy = max(0, min(31, (int)floorf(y * 32.0f)));
  int cz = max(0, min(31, (int)floorf(z * 32.0f)));
  return (spread3((unsigned)cx) << 2) | (spread3((unsigned)cy) << 1) |
         spread3((unsigned)cz);
}

extern "C" __global__ void __launch_bounds__(256)
msgrid_hist_kernel(const float* __restrict__ xyz, unsigned* __restrict__ hist,
                   int npts) {
  int n = blockIdx.x * 256 + threadIdx.x;
  if (n >= npts) return;
  float x = xyz[3 * n + 0], y = xyz[3 * n + 1], z = xyz[3 * n + 2];
  atomicAdd(&hist[bin_key(x, y, z)], 1u);
}

// Single-block exclusive prefix sum over NBINS counters (in place).
extern "C" __global__ void __launch_bounds__(SCAN_THREADS)
msgrid_scan_kernel(unsigned* __restrict__ hist) {
  __shared__ unsigned part[SCAN_THREADS];
  int t = threadIdx.x;
  unsigned vals[NBINS / SCAN_THREADS];
  unsigned sum = 0;
#pragma unroll
  for (int i = 0; i < NBINS / SCAN_THREADS; ++i) {
    vals[i] = hist[t * (NBINS / SCAN_THREADS) + i];
    sum += vals[i];
  }
  part[t] = sum;
  __syncthreads();
  for (int off = 1; off < SCAN_THREADS; off <<= 1) {
    unsigned v = (t >= off) ? part[t - off] : 0u;
    __syncthreads();
    part[t] += v;
    __syncthreads();
  }
  unsigned run = (t == 0) ? 0u : part[t - 1];   // exclusive base
#pragma unroll
  for (int i = 0; i < NBINS / SCAN_THREADS; ++i) {
    hist[t * (NBINS / SCAN_THREADS) + i] = run;
    run += vals[i];
  }
}

extern "C" __global__ void __launch_bounds__(256)
msgrid_scatter_kernel(const float* __restrict__ xyz,
                      unsigned* __restrict__ offsets,
                      unsigned* __restrict__ perm,
                      float* __restrict__ sxyz, int npts) {
  int n = blockIdx.x * 256 + threadIdx.x;
  if (n >= npts) return;
  float x = xyz[3 * n + 0], y = xyz[3 * n + 1], z = xyz[3 * n + 2];
  unsigned pos = atomicAdd(&offsets[bin_key(x, y, z)], 1u);
  perm[pos] = (unsigned)n;
  sxyz[3 * pos + 0] = x;
  sxyz[3 * pos + 1] = y;
  sxyz[3 * pos + 2] = z;
}

// ------------------------------------------------------------ main gathers --
extern "C" __global__ void __launch_bounds__(256)
msgrid_gather_sorted_kernel(const float* __restrict__ sxyz,
                            const unsigned* __restrict__ perm,
                            const float* __restrict__ g0,
                            const float* __restrict__ g1,
                            const float* __restrict__ g2,
                            float* __restrict__ out, int npts) {
  int i = blockIdx.x * 256 + threadIdx.x;
  if (i >= npts) return;

  // Prefetch the upcoming sorted-xyz stream (global_prefetch_b8).
  __builtin_prefetch(sxyz + 3 * (i + 16384), 0, 3);

  int n = (int)perm[i];
  float x = __builtin_nontemporal_load(sxyz + 3 * i + 0);
  float y = __builtin_nontemporal_load(sxyz + 3 * i + 1);
  float z = __builtin_nontemporal_load(sxyz + 3 * i + 2);

  float acc[NCH];
  sample_all(x, y, z, g0, g1, g2, acc, /*nt2=*/false);
  store_out(out, n, acc);
}

extern "C" __global__ void __launch_bounds__(256)
msgrid_direct_kernel(const float* __restrict__ xyz,
                     const float* __restrict__ g0,
                     const float* __restrict__ g1,
                     const float* __restrict__ g2,
                     float* __restrict__ out, int npts) {
  int n = blockIdx.x * 256 + threadIdx.x;
  if (n >= npts) return;
  __builtin_prefetch(xyz + 3 * (n + 16384), 0, 3);
  float x = __builtin_nontemporal_load(xyz + 3 * n + 0);
  float y = __builtin_nontemporal_load(xyz + 3 * n + 1);
  float z = __builtin_nontemporal_load(xyz + 3 * n + 2);
  float acc[NCH];
  sample_all(x, y, z, g0, g1, g2, acc, /*nt2=*/true);
  store_out(out, n, acc);
}

// ------------------------------------------------------------------ launch --
extern "C" void kernel_launch(void* const* d_in, const int* in_sizes, int n_in,
                              void* d_out, int out_size, void* d_ws, size_t ws_size,
                              hipStream_t stream) {
  const float* xyz = (const float*)d_in[0];
  const float* g0  = (const float*)d_in[1];
  const float* g1  = (const float*)d_in[2];
  const float* g2  = (const float*)d_in[3];
  float* out = (float*)d_out;

  int npts = in_sizes[0] / 3;
  int blocks = (npts + 255) / 256;

  size_t histB = (size_t)NBINS * 4;
  size_t permB = (size_t)npts * 4;
  size_t sxyzB = (size_t)npts * 12;
  size_t need = histB + permB + sxyzB;

  if (ws_size >= need) {
    unsigned* hist = (unsigned*)d_ws;
    unsigned* perm = (unsigned*)((char*)d_ws + histB);
    float* sxyz    = (float*)((char*)d_ws + histB + permB);

    hipMemsetAsync(hist, 0, histB, stream);
    msgrid_hist_kernel<<<blocks, 256, 0, stream>>>(xyz, hist, npts);
    msgrid_scan_kernel<<<1, SCAN_THREADS, 0, stream>>>(hist);
    msgrid_scatter_kernel<<<blocks, 256, 0, stream>>>(xyz, hist, perm, sxyz,
                                                      npts);
    msgrid_gather_sorted_kernel<<<blocks, 256, 0, stream>>>(sxyz, perm, g0, g1,
                                                            g2, out, npts);
  } else {
    msgrid_direct_kernel<<<blocks, 256, 0, stream>>>(xyz, g0, g1, g2, out,
                                                     npts);
  }
}